// AFGCN_18030272708968
// MI455X (gfx1250) — compile-verified
//
#include <hip/hip_runtime.h>

typedef __attribute__((ext_vector_type(2))) float v2f;
typedef __attribute__((ext_vector_type(8))) float v8f;

#define D_FEAT 128

// ---------------------------------------------------------------------------
// 1) zero the aggregation buffer
// ---------------------------------------------------------------------------
__global__ __launch_bounds__(256) void afgcn_zero(float4* __restrict__ p, int n4) {
    int i = blockIdx.x * blockDim.x + threadIdx.x;
    if (i < n4) p[i] = make_float4(0.f, 0.f, 0.f, 0.f);
}

// ---------------------------------------------------------------------------
// 2) edge scatter-add: one wave32 per edge, lane handles 4 floats of the row
// ---------------------------------------------------------------------------
__global__ __launch_bounds__(256) void afgcn_scatter(const float* __restrict__ h,
                                                     const int* __restrict__ src,
                                                     const int* __restrict__ dst,
                                                     float* __restrict__ agg,
                                                     int nedges) {
    int gid  = blockIdx.x * blockDim.x + threadIdx.x;
    int e    = gid >> 5;
    int lane = gid & 31;
    if (e >= nedges) return;
    int s = src[e];
    int d = dst[e];
    const float4 v = *(const float4*)(h + (size_t)s * D_FEAT + lane * 4);
    float* ap = agg + (size_t)d * D_FEAT + lane * 4;
    atomicAdd(ap + 0, v.x);
    atomicAdd(ap + 1, v.y);
    atomicAdd(ap + 2, v.z);
    atomicAdd(ap + 3, v.w);
}

// ---------------------------------------------------------------------------
// One GEMM accumulation phase: acc += A_tile(16x128) @ W(128x128).
// Force-inlined with concrete pointer args so A loads stay in the GLOBAL
// address space (global_load_b64, LOADcnt only) instead of degrading to
// FLAT (which couples LOADcnt+DScnt and serializes against the LDS pipe).
// ---------------------------------------------------------------------------
__device__ __forceinline__ void gemm_phase(const float* __restrict__ A,
                                           const float* __restrict__ W,
                                           float2* __restrict__ wfrag,
                                           v8f acc[8], int tid, int arow,
                                           int half, int laneM) {
    __syncthreads();
    // stage W^T fragments: wfrag[kp*128+col] = {W[2kp][col], W[2kp+1][col]}
    for (int i = tid; i < 64 * D_FEAT; i += 256) {
        int kp  = i >> 7;
        int col = i & 127;
        wfrag[i] = make_float2(W[(2 * kp) * D_FEAT + col],
                               W[(2 * kp + 1) * D_FEAT + col]);
    }
    __syncthreads();

    const float* aptr = A + (size_t)arow * D_FEAT + half * 2;
#pragma unroll
    for (int kb = 0; kb < 32; ++kb) {
        // A 16x4 frag: lanes 0-15 -> K=k0,k0+1 ; lanes 16-31 -> K=k0+2,k0+3
        float2 af = *(const float2*)(aptr + kb * 4);
        v2f a;
        a.x = af.x;
        a.y = af.y;
#pragma unroll
        for (int n0 = 0; n0 < 8; ++n0) {
            // B 4x16 frag, same K split, N = laneM (+16*n0 column tile)
            float2 bf = wfrag[(kb * 2 + half) * D_FEAT + n0 * 16 + laneM];
            v2f b;
            b.x = bf.x;
            b.y = bf.y;
            acc[n0] = __builtin_amdgcn_wmma_f32_16x16x4_f32(
                    false, a, false, b, (short)0, acc[n0], false, false);
        }
    }
}

// ---------------------------------------------------------------------------
// 3) h_out = relu(agg @ Wn + h_in @ Wr + b) + x0      (one layer)
//    256 threads = 8 waves; each wave computes a 16x128 output tile with
//    V_WMMA_F32_16X16X4_F32 (fp32 in, fp32 accum -> matches reference math).
// ---------------------------------------------------------------------------
__global__ __launch_bounds__(256) void afgcn_gemm_relu_res(
        const float* __restrict__ agg, const float* __restrict__ hin,
        const float* __restrict__ x0,  const float* __restrict__ Wn,
        const float* __restrict__ Wr,  const float* __restrict__ bias,
        float* __restrict__ hout, int nrows) {
    __shared__ float2 wfrag[64 * D_FEAT];   // 64 KB: [kpair 0..63][col 0..127]

    const int tid     = threadIdx.x;
    const int lane    = tid & 31;
    const int wid     = tid >> 5;                   // 0..7
    const int rowBase = blockIdx.x * 128 + wid * 16;
    const int laneM   = lane & 15;                  // M (A) or N (B,C/D) index
    const int half    = lane >> 4;                  // K sub-pair selector

    v8f acc[8];
    const v8f zero8 = {0.f, 0.f, 0.f, 0.f, 0.f, 0.f, 0.f, 0.f};
#pragma unroll
    for (int i = 0; i < 8; ++i) acc[i] = zero8;

    // A row, clamped via select so EXEC stays all-ones for WMMA
    int arow = rowBase + laneM;
    arow = (arow < nrows) ? arow : 0;

    gemm_phase(agg, Wn, wfrag, acc, tid, arow, half, laneM);  // agg @ Wn
    gemm_phase(hin, Wr, wfrag, acc, tid, arow, half, laneM);  // + h @ Wr

    // epilogue: bias + relu + residual, per documented C/D layout
    float bv[8];
#pragma unroll
    for (int n0 = 0; n0 < 8; ++n0) bv[n0] = bias[n0 * 16 + laneM];

#pragma unroll
    for (int v = 0; v < 8; ++v) {
        int row = rowBase + v + half * 8;   // M = v (lanes 0-15), v+8 (16-31)
        if (row < nrows) {
#pragma unroll
            for (int n0 = 0; n0 < 8; ++n0) {
                int col   = n0 * 16 + laneM;
                float val = acc[n0][v] + bv[n0];
                val = fmaxf(val, 0.f) + x0[(size_t)row * D_FEAT + col];
                hout[(size_t)row * D_FEAT + col] = val;
            }
        }
    }
}

// ---------------------------------------------------------------------------
// 4) out[n] = dot(h[n], fcW) + fcb    (one wave32 per node)
// ---------------------------------------------------------------------------
__global__ __launch_bounds__(256) void afgcn_fc(const float* __restrict__ h,
                                                const float* __restrict__ w,
                                                const float* __restrict__ bptr,
                                                float* __restrict__ out, int n) {
    int gid  = blockIdx.x * blockDim.x + threadIdx.x;
    int node = gid >> 5;
    int lane = gid & 31;
    if (node >= n) return;
    const float4 hv = *(const float4*)(h + (size_t)node * D_FEAT + lane * 4);
    const float4 wv = *(const float4*)(w + lane * 4);
    float p = hv.x * wv.x + hv.y * wv.y + hv.z * wv.z + hv.w * wv.w;
#pragma unroll
    for (int o = 16; o > 0; o >>= 1) p += __shfl_xor(p, o, 32);
    if (lane == 0) out[node] = p + bptr[0];
}

// ---------------------------------------------------------------------------
extern "C" void kernel_launch(void* const* d_in, const int* in_sizes, int n_in,
                              void* d_out, int out_size, void* d_ws, size_t ws_size,
                              hipStream_t stream) {
    const int D = D_FEAT;
    const int N = in_sizes[0] / D;            // 50000
    const int E = in_sizes[1] / 2;            // 600000
    const int L = in_sizes[2] / (D * D);      // 4

    const float* x   = (const float*)d_in[0];
    const int*   ei  = (const int*)d_in[1];
    const float* Wn  = (const float*)d_in[2];
    const float* Wr  = (const float*)d_in[3];
    const float* b   = (const float*)d_in[4];
    const float* fcW = (const float*)d_in[5];
    const float* fcb = (const float*)d_in[6];
    float*       out = (float*)d_out;

    // workspace carve: agg | hA | hB  (each N*D floats)
    float* agg = (float*)d_ws;
    float* hA  = agg + (size_t)N * D;
    float* hB  = hA + (size_t)N * D;

    const int* src = ei;
    const int* dst = ei + E;

    const float* h = x;
    for (int l = 0; l < L; ++l) {
        int n4 = (N * D) / 4;
        afgcn_zero<<<(n4 + 255) / 256, 256, 0, stream>>>((float4*)agg, n4);

        long long threads = (long long)E * 32;
        afgcn_scatter<<<(int)((threads + 255) / 256), 256, 0, stream>>>(
                h, src, dst, agg, E);

        float* hout = (l & 1) ? hB : hA;
        afgcn_gemm_relu_res<<<(N + 127) / 128, 256, 0, stream>>>(
                agg, h, x, Wn + (size_t)l * D * D, Wr + (size_t)l * D * D,
                b + (size_t)l * D, hout, N);
        h = hout;
    }

    long long fcth = (long long)N * 32;
    afgcn_fc<<<(int)((fcth + 255) / 256), 256, 0, stream>>>(h, fcW, fcb, out, N);
}